// PageManager_3693671874796
// MI455X (gfx1250) — compile-verified
//
#include <hip/hip_runtime.h>
#include <hip/hip_bf16.h>
#include <stdint.h>

// Problem constants (from reference setup_inputs)
#define H_HEADS     8
#define D_DIM       128
#define S_LEN       4096
#define NUM_PAGES   4096
#define PAGE_SIZE   16
#define MAX_SEQS    8
#define MAX_PPS     256
#define PAGE_ELEMS  (PAGE_SIZE * D_DIM)                      // 2048 floats = 8 KB
#define PAGES_ELEMS ((size_t)2 * H_HEADS * NUM_PAGES * PAGE_ELEMS)  // 134,217,728 floats

// Native clang vector (HIP's float4 is a struct and is rejected by the
// nontemporal builtins).
typedef float v4f __attribute__((ext_vector_type(4)));

// ---------------------------------------------------------------------------
// Kernel 1: first-fit page allocation (parallel prefix scan over free mask).
// Produces:
//   map[p]  = block index b if page p is allocated this step, else -1
//   out_pi  = updated page_indices as float
//   out_seq = updated seq_page_indices as float
// ---------------------------------------------------------------------------
__global__ __launch_bounds__(256) void pm_alloc_kernel(
    const int* __restrict__ page_indices,
    const int* __restrict__ seq_page_indices,
    const int* __restrict__ slot_p,
    const int* __restrict__ tl_p,
    int*   __restrict__ map,
    float* __restrict__ out_pi,
    float* __restrict__ out_seq) {
  __shared__ int scan[256];
  const int t    = threadIdx.x;
  const int slot = slot_p[0];
  const int nb   = (tl_p[0] + PAGE_SIZE - 1) / PAGE_SIZE;   // ceil(true_length / page_size)

  // Copy seq_page_indices input -> output (overwritten below for slot row).
  for (int i = t; i < MAX_SEQS * MAX_PPS; i += 256)
    out_seq[i] = (float)seq_page_indices[i];

  // Each thread owns 16 consecutive pages; count free ones (page 0 reserved).
  int pi[16];
  int cnt = 0;
#pragma unroll
  for (int j = 0; j < 16; ++j) {
    const int p = t * 16 + j;
    pi[j] = page_indices[p];
    cnt += (p > 0 && pi[j] == 0) ? 1 : 0;
  }
  scan[t] = cnt;

  // Hillis-Steele inclusive scan over 256 partial counts.
  for (int off = 1; off < 256; off <<= 1) {
    __syncthreads();
    const int add = (t >= off) ? scan[t - off] : 0;
    __syncthreads();
    scan[t] += add;
  }
  __syncthreads();
  int r = scan[t] - cnt;  // exclusive prefix: rank of first free page in my range

#pragma unroll
  for (int j = 0; j < 16; ++j) {
    const int p       = t * 16 + j;
    const bool isfree = (p > 0) && (pi[j] == 0);
    const bool take   = isfree && (r < nb);
    map[p]    = take ? r : -1;
    out_pi[p] = take ? 1.0f : (float)pi[j];
    if (take) out_seq[slot * MAX_PPS + r] = (float)p;
    if (isfree) ++r;
  }
}

// ---------------------------------------------------------------------------
// Kernel 2: single pass over the [2, H, NUM_PAGES, PAGE_SIZE, D] output.
// One workgroup per 8 KB page. If the page was allocated this step, gather
// the block from key/value (strided, NT B128). Otherwise it is a straight
// page copy routed through LDS with CDNA5 async global<->LDS DMA ops.
// ---------------------------------------------------------------------------
__global__ __launch_bounds__(256) void pm_merge_copy_kernel(
    const float* __restrict__ key,
    const float* __restrict__ value,
    const float* __restrict__ key_pages,
    const float* __restrict__ value_pages,
    const int*   __restrict__ map,
    float* __restrict__ out_pages) {
  __shared__ __align__(16) float smem[PAGE_ELEMS];  // 8 KB staging
  const int q  = blockIdx.x;                 // kv*32768 + h*4096 + p
  const int p  = q & (NUM_PAGES - 1);
  const int h  = (q >> 12) & (H_HEADS - 1);
  const int kv = q >> 15;
  const int t  = threadIdx.x;
  const int b  = map[p];                     // uniform per block
  float* dst = out_pages + (size_t)q * PAGE_ELEMS;

  if (b >= 0) {
    // Scatter path: pull rows of this block from key/value [1,S,H,D].
    const float* src = (kv == 0) ? key : value;
#pragma unroll
    for (int c = 0; c < 2; ++c) {
      const int e = (t + c * 256) * 4;       // element offset within page
      const int r = e >> 7;                  // row within page (0..15)
      const int d = e & (D_DIM - 1);         // 0..124, float4-aligned
      const v4f v4 = __builtin_nontemporal_load(
          (const v4f*)(src + ((size_t)((b * PAGE_SIZE + r) * H_HEADS + h)) * D_DIM + d));
      __builtin_nontemporal_store(v4, (v4f*)(dst + e));
    }
  } else {
    // Copy path: async DMA global -> LDS -> global (B128 per lane, NT).
    const float* src = ((kv == 0) ? key_pages : value_pages)
                     + (size_t)(h * NUM_PAGES + p) * PAGE_ELEMS;
    const uint32_t lds_base = (uint32_t)(uintptr_t)(&smem[0]);
#pragma unroll
    for (int c = 0; c < 2; ++c) {
      const uint32_t off      = (uint32_t)((t + c * 256) * 16);  // bytes
      const uint32_t lds_addr = lds_base + off;
      const uint64_t gsrc     = (uint64_t)(uintptr_t)src + off;
      asm volatile("global_load_async_to_lds_b128 %0, %1, off th:TH_LOAD_NT"
                   :: "v"(lds_addr), "v"(gsrc) : "memory");
    }
    asm volatile("s_wait_asynccnt 0" ::: "memory");
#pragma unroll
    for (int c = 0; c < 2; ++c) {
      const uint32_t off      = (uint32_t)((t + c * 256) * 16);
      const uint32_t lds_addr = lds_base + off;
      const uint64_t gdst     = (uint64_t)(uintptr_t)dst + off;
      asm volatile("global_store_async_from_lds_b128 %0, %1, off th:TH_STORE_NT"
                   :: "v"(gdst), "v"(lds_addr) : "memory");
    }
    asm volatile("s_wait_asynccnt 0" ::: "memory");
  }
}

// ---------------------------------------------------------------------------
extern "C" void kernel_launch(void* const* d_in, const int* in_sizes, int n_in,
                              void* d_out, int out_size, void* d_ws, size_t ws_size,
                              hipStream_t stream) {
  (void)in_sizes; (void)n_in; (void)out_size; (void)ws_size;

  const float* key          = (const float*)d_in[0];  // [1,4096,8,128]
  const float* value        = (const float*)d_in[1];  // [1,4096,8,128]
  const float* key_pages    = (const float*)d_in[2];  // [8,4096,16,128]
  const float* value_pages  = (const float*)d_in[3];  // [8,4096,16,128]
  const int*   page_indices = (const int*)d_in[4];    // [4096]
  const int*   seq_page_idx = (const int*)d_in[5];    // [8,256]
  const int*   slot_p       = (const int*)d_in[6];    // scalar
  const int*   tl_p         = (const int*)d_in[7];    // scalar

  float* out       = (float*)d_out;
  float* out_pages = out;                       // [2,8,4096,16,128]
  float* out_pi    = out + PAGES_ELEMS;         // [4096]
  float* out_seq   = out_pi + NUM_PAGES;        // [8,256]
  int*   map       = (int*)d_ws;                // [4096] page -> block or -1

  pm_alloc_kernel<<<dim3(1), dim3(256), 0, stream>>>(
      page_indices, seq_page_idx, slot_p, tl_p, map, out_pi, out_seq);

  pm_merge_copy_kernel<<<dim3(2 * H_HEADS * NUM_PAGES), dim3(256), 0, stream>>>(
      key, value, key_pages, value_pages, map, out_pages);
}